// MolhivNet_20143396618972
// MI455X (gfx1250) — compile-verified
//
#include <hip/hip_runtime.h>

#define NN 10000          // N_NODES
#define NE 160000         // N_EDGES
#define EDGE_DIM 3
#define LIN 32
#define HID 300
#define PAD 320           // padded K/N (10 chunks of 32, 20 tiles of 16)
#define NLAYERS 5
#define NT 5              // N-tiles (of 16) per wave: 16x80 output strip

typedef float v2f __attribute__((ext_vector_type(2)));
typedef float v8f __attribute__((ext_vector_type(8)));

// ---------------------------------------------------------------- utilities
__global__ void fill_f32(float* __restrict__ p, float v, int n) {
  int i = blockIdx.x * blockDim.x + threadIdx.x;
  if (i < n) p[i] = v;
}

// zero-padded weight copy: wp[Kp,PAD] <- w[Ksrc,HID]
__global__ void pad_weight(const float* __restrict__ w, float* __restrict__ wp,
                           int Ksrc, int Kp) {
  int tid = blockIdx.x * blockDim.x + threadIdx.x;
  if (tid >= Kp * PAD) return;
  int k = tid / PAD, n = tid % PAD;
  wp[tid] = (k < Ksrc && n < HID) ? w[k * HID + n] : 0.0f;
}

// e[E,32] = edge_attr[E,3] @ edge_w[3,32] + edge_b   (row stride 32 = K of layer 0)
__global__ void edge_embed(const float* __restrict__ ea, const float* __restrict__ w,
                           const float* __restrict__ b, float* __restrict__ e) {
  int tid = blockIdx.x * blockDim.x + threadIdx.x;
  if (tid >= NE * LIN) return;
  int ed = tid / LIN, j = tid % LIN;
  const float* a = ea + (size_t)ed * EDGE_DIM;
  float v = b[j];
  v = fmaf(a[0], w[0 * LIN + j], v);
  v = fmaf(a[1], w[1 * LIN + j], v);
  v = fmaf(a[2], w[2 * LIN + j], v);
  e[tid] = v;
}

// deg[col[e]] += 1  (col values < NN)
__global__ void deg_count(const int* __restrict__ ei, float* __restrict__ deg) {
  int e = blockIdx.x * blockDim.x + threadIdx.x;
  if (e >= NE) return;
  __hip_atomic_fetch_add(&deg[ei[NE + e]], 1.0f, __ATOMIC_RELAXED,
                         __HIP_MEMORY_SCOPE_AGENT);
}

__global__ void compute_dinv(const float* __restrict__ deg, float* __restrict__ dinv) {
  int i = blockIdx.x * blockDim.x + threadIdx.x;
  if (i < NN) dinv[i] = rsqrtf(deg[i] + 1.0f);
}

__global__ void compute_norm(const int* __restrict__ ei, const float* __restrict__ dinv,
                             float* __restrict__ norm) {
  int e = blockIdx.x * blockDim.x + threadIdx.x;
  if (e >= NE) return;
  norm[e] = dinv[ei[e]] * dinv[ei[NE + e]];
}

// ------------------------------------------------------ fp32 WMMA GEMM
// C[M,PAD] = A[M,KP] @ Bp[KP,PAD]; compile-time padded dims -> no guards.
// block = 256 threads = 8 waves; wave w -> M-tile (blockIdx.x*8+w)*16,
// each wave computes NT=5 N-tiles (16x80 strip) so every A load feeds 5 WMMAs.
// B chunk (32 x 80) staged in LDS *pair-interleaved*: the {k,k+1} values of a
// column are contiguous 8B -> each B operand is one aligned ds_load_b64.
//   Bs[((p*NT + t)*16 + l)*2 + (k&1)],  p = k_local/2
template <int KP>
__global__ __launch_bounds__(256)
void gemm_f32_wmma(const float* __restrict__ A, const float* __restrict__ Bp,
                   float* __restrict__ C) {
  constexpr int NW = 16 * NT;               // 80 cols per wave strip
  __shared__ float Bs[32 * NW];             // 10 KB
  const int lane = threadIdx.x & 31;
  const int wave = threadIdx.x >> 5;
  const int half = lane >> 4;   // 0: K pair {k,k+1}, 1: K pair {k+2,k+3}
  const int l16  = lane & 15;
  const int m0 = (blockIdx.x * 8 + wave) * 16;
  const int n0 = blockIdx.y * NW;
  const float* __restrict__ arow = A + (size_t)(m0 + l16) * KP;

  v8f c[NT];
#pragma unroll
  for (int t = 0; t < NT; ++t) c[t] = (v8f){};

  for (int kc = 0; kc < KP; kc += 32) {
    // stage 32 x 80 B chunk, pair-interleaved: 2560 elems / 256 threads
#pragma unroll
    for (int q = 0; q < (32 * NW) / 256; ++q) {
      int idx = threadIdx.x + q * 256;
      int kr = idx / NW, cn = idx % NW;
      int t = cn >> 4, l = cn & 15, p = kr >> 1, par = kr & 1;
      Bs[((p * NT + t) * 16 + l) * 2 + par] = Bp[(size_t)(kc + kr) * PAD + n0 + cn];
    }
    __syncthreads();
#pragma unroll
    for (int kk = 0; kk < 32; kk += 4) {
      const int kg = kc + kk + half * 2;         // even, 8B-aligned from arow
      v2f a = *(const v2f*)(arow + kg);          // global_load_b64, reused NT x
      const int p = (kk >> 1) + half;            // K-pair index in chunk
#pragma unroll
      for (int t = 0; t < NT; ++t) {
        v2f b = *(const v2f*)&Bs[((p * NT + t) * 16 + l16) * 2];  // ds_load_b64
        c[t] = __builtin_amdgcn_wmma_f32_16x16x4_f32(false, a, false, b,
                                                     (short)0, c[t], false, false);
      }
    }
    __syncthreads();
  }

#pragma unroll
  for (int t = 0; t < NT; ++t)
#pragma unroll
    for (int r = 0; r < 8; ++r)
      C[(size_t)(m0 + r + 8 * half) * PAD + n0 + t * 16 + l16] = c[t][r];
}

// agg[col[e], d] += hw[row[e], d] * norm[e]   (buffers have row stride PAD)
__global__ void scatter_agg(const float* __restrict__ hw, const int* __restrict__ ei,
                            const float* __restrict__ norm, float* __restrict__ agg) {
  long long tid = (long long)blockIdx.x * blockDim.x + threadIdx.x;
  if (tid >= (long long)NE * HID) return;
  int e = (int)(tid / HID);
  int d = (int)(tid % HID);
  int r = ei[e];
  int c = ei[NE + e];
  float v = hw[(size_t)r * PAD + d] * norm[e];
  __hip_atomic_fetch_add(&agg[(size_t)c * PAD + d], v, __ATOMIC_RELAXED,
                         __HIP_MEMORY_SCOPE_AGENT);
}

// h = relu(agg + self_norm * hw + bias), in place; pad columns stay 0
__global__ void finalize_layer(float* __restrict__ h, const float* __restrict__ agg,
                               const float* __restrict__ dinv,
                               const float* __restrict__ bias) {
  long long tid = (long long)blockIdx.x * blockDim.x + threadIdx.x;
  if (tid >= (long long)NE * HID) return;
  int i = (int)(tid / HID);
  int d = (int)(tid % HID);
  size_t idx = (size_t)i * PAD + d;
  float hw = h[idx];
  float sn, ag;
  if (i < NN) {
    float di = dinv[i];
    sn = di * di;
    ag = agg[(size_t)i * PAD + d];
  } else {        // empty segments: deg=1 -> self_norm=1, agg=0
    sn = 1.0f;
    ag = 0.0f;
  }
  h[idx] = fmaxf(fmaf(sn, hw, ag) + bias[d], 0.0f);
}

// column sums of h[E,PAD] into gsum[PAD] (pad cols are zero; contiguous reads)
__global__ __launch_bounds__(256)
void col_sum(const float* __restrict__ h, float* __restrict__ gsum) {
  __shared__ float ls[PAD];
  for (int d = threadIdx.x; d < PAD; d += blockDim.x) ls[d] = 0.0f;
  __syncthreads();
  const long long total = (long long)NE * PAD;
  for (long long idx = (long long)blockIdx.x * blockDim.x + threadIdx.x; idx < total;
       idx += (long long)gridDim.x * blockDim.x) {
    int d = (int)(idx % PAD);
    __hip_atomic_fetch_add(&ls[d], h[idx], __ATOMIC_RELAXED,
                           __HIP_MEMORY_SCOPE_WORKGROUP);
  }
  __syncthreads();
  for (int d = threadIdx.x; d < PAD; d += blockDim.x)
    __hip_atomic_fetch_add(&gsum[d], ls[d], __ATOMIC_RELAXED,
                           __HIP_MEMORY_SCOPE_AGENT);
}

// mean -> lin1(relu) -> lin2 ; single block
__global__ void head(const float* __restrict__ gsum, const float* __restrict__ w1,
                     const float* __restrict__ b1, const float* __restrict__ w2,
                     const float* __restrict__ b2, float* __restrict__ out) {
  __shared__ float gm[HID];
  __shared__ float s[LIN];
  const int t = threadIdx.x;
  for (int d = t; d < HID; d += blockDim.x) gm[d] = gsum[d] * (1.0f / (float)NE);
  __syncthreads();
  if (t < LIN) {
    float v = b1[t];
    for (int d = 0; d < HID; ++d) v = fmaf(gm[d], w1[d * LIN + t], v);
    s[t] = fmaxf(v, 0.0f);
  }
  __syncthreads();
  if (t < 2) {
    float v = b2[t];
    for (int j = 0; j < LIN; ++j) v = fmaf(s[j], w2[j * 2 + t], v);
    out[t] = v;
  }
}

// ---------------------------------------------------------------- launch
extern "C" void kernel_launch(void* const* d_in, const int* in_sizes, int n_in,
                              void* d_out, int out_size, void* d_ws, size_t ws_size,
                              hipStream_t stream) {
  (void)in_sizes; (void)n_in; (void)out_size; (void)ws_size;
  // inputs (setup_inputs order)
  const int*   edge_index = (const int*)  d_in[1];   // [2, NE]
  const float* edge_attr  = (const float*)d_in[2];   // [NE, 3]
  const float* edge_w     = (const float*)d_in[6];   // [3, 32]
  const float* edge_b     = (const float*)d_in[7];   // [32]
  const float* gcn0_w     = (const float*)d_in[8];   // [32, 300]
  const float* gcn0_b     = (const float*)d_in[9];   // [300]
  const float* gcn_w      = (const float*)d_in[10];  // [4, 300, 300]
  const float* gcn_b      = (const float*)d_in[11];  // [4, 300]
  const float* lin1_w     = (const float*)d_in[12];  // [300, 32]
  const float* lin1_b     = (const float*)d_in[13];  // [32]
  const float* lin2_w     = (const float*)d_in[14];  // [32, 2]
  const float* lin2_b     = (const float*)d_in[15];  // [2]
  float* out = (float*)d_out;

  // workspace carve-up (floats)
  float* ws = (float*)d_ws;
  const size_t HP = (size_t)NE * PAD;               // 51,200,000
  float* bufA  = ws;                                // [NE, PAD]
  float* bufB  = bufA + HP;                         // [NE, PAD]
  float* agg   = bufB + HP;                         // [NN, PAD]
  float* ebuf  = agg + (size_t)NN * PAD;            // [NE, 32]
  float* wpad0 = ebuf + (size_t)NE * LIN;           // [32, PAD]
  float* wpad  = wpad0 + (size_t)LIN * PAD;         // [4, PAD, PAD]
  float* deg   = wpad + (size_t)4 * PAD * PAD;      // [NN]
  float* dinv  = deg + NN;                          // [NN]
  float* norm  = dinv + NN;                         // [NE]
  float* gsum  = norm + NE;                         // [PAD]

  const int B256 = 256;
  const dim3 gemm_grid(NE / 128, PAD / (16 * NT)); // 1250 x 4
  const int eh_blocks  = (NE * LIN + B256 - 1) / B256;
  const int eE_blocks  = (NE + B256 - 1) / B256;
  const int nN_blocks  = (NN + B256 - 1) / B256;
  const long long EH   = (long long)NE * HID;
  const int EH_blocks  = (int)((EH + B256 - 1) / B256);
  const int agg_blocks = (NN * PAD + B256 - 1) / B256;
  const int w0_blocks  = (LIN * PAD + B256 - 1) / B256;
  const int wl_blocks  = (PAD * PAD + B256 - 1) / B256;

  // 0) pad weights into workspace (zero-fill beyond [K,300])
  pad_weight<<<w0_blocks, B256, 0, stream>>>(gcn0_w, wpad0, LIN, LIN);
  for (int l = 0; l < NLAYERS - 1; ++l)
    pad_weight<<<wl_blocks, B256, 0, stream>>>(gcn_w + (size_t)l * HID * HID,
                                               wpad + (size_t)l * PAD * PAD,
                                               HID, PAD);

  // 1) edge embeddings (h0 = e, row stride 32)
  edge_embed<<<eh_blocks, B256, 0, stream>>>(edge_attr, edge_w, edge_b, ebuf);

  // 2) gcn_norm: deg, dinv, per-edge norm
  fill_f32<<<nN_blocks, B256, 0, stream>>>(deg, 0.0f, NN);
  deg_count<<<eE_blocks, B256, 0, stream>>>(edge_index, deg);
  compute_dinv<<<nN_blocks, B256, 0, stream>>>(deg, dinv);
  compute_norm<<<eE_blocks, B256, 0, stream>>>(edge_index, dinv, norm);

  // 3) layer 0: hw = e @ W0 ; aggregate ; relu-finalize
  gemm_f32_wmma<LIN><<<gemm_grid, B256, 0, stream>>>(ebuf, wpad0, bufA);
  fill_f32<<<agg_blocks, B256, 0, stream>>>(agg, 0.0f, NN * PAD);
  scatter_agg<<<EH_blocks, B256, 0, stream>>>(bufA, edge_index, norm, agg);
  finalize_layer<<<EH_blocks, B256, 0, stream>>>(bufA, agg, dinv, gcn0_b);

  // 4) layers 1..4 (ping-pong bufA/bufB)
  float* cur = bufA;
  float* nxt = bufB;
  for (int l = 1; l < NLAYERS; ++l) {
    const float* Wp = wpad + (size_t)(l - 1) * PAD * PAD;
    const float* b  = gcn_b + (size_t)(l - 1) * HID;
    gemm_f32_wmma<PAD><<<gemm_grid, B256, 0, stream>>>(cur, Wp, nxt);
    fill_f32<<<agg_blocks, B256, 0, stream>>>(agg, 0.0f, NN * PAD);
    scatter_agg<<<EH_blocks, B256, 0, stream>>>(nxt, edge_index, norm, agg);
    finalize_layer<<<EH_blocks, B256, 0, stream>>>(nxt, agg, dinv, b);
    float* t = cur; cur = nxt; nxt = t;
  }

  // 5) graph readout + MLP head
  fill_f32<<<(PAD + B256 - 1) / B256, B256, 0, stream>>>(gsum, 0.0f, PAD);
  col_sum<<<2048, B256, 0, stream>>>(cur, gsum);
  head<<<1, 320, 0, stream>>>(gsum, lin1_w, lin1_b, lin2_w, lin2_b, out);
}